// SAGEConv_2293512536931
// MI455X (gfx1250) — compile-verified
//
#include <hip/hip_runtime.h>

typedef float v2f __attribute__((ext_vector_type(2)));
typedef float v8f __attribute__((ext_vector_type(8)));

#define N_NODES 100000
#define N_EDGES 1250000
#define FEAT 64           // in-features = out-features = 64, K = 2*FEAT = 128

// ---------------------------------------------------------------------------
// Kernel 1: zero the workspace accumulators (msum [N,64] and deg [N])
// ---------------------------------------------------------------------------
__global__ void sage_zero_kernel(float* __restrict__ ws, int n) {
    int i = blockIdx.x * blockDim.x + threadIdx.x;
    int stride = gridDim.x * blockDim.x;
    for (; i < n; i += stride) ws[i] = 0.0f;
}

// ---------------------------------------------------------------------------
// Kernel 2: edge scatter.  16 threads per edge, each handles a float4 of the
// 64-float feature row: gather h[src] (coalesced float4) and atomic-add into
// msum[dst].  Lane 0 of each edge group bumps the degree counter.
// msum/deg total ~26MB -> resident in 192MB L2, atomics resolve in L2.
// ---------------------------------------------------------------------------
__global__ void sage_edge_scatter_kernel(const float* __restrict__ h,
                                         const int*   __restrict__ src,
                                         const int*   __restrict__ dst,
                                         float* __restrict__ msum,
                                         float* __restrict__ deg) {
    long long idx = (long long)blockIdx.x * blockDim.x + threadIdx.x;
    long long e   = idx >> 4;          // edge id
    int       q   = (int)(idx & 15);   // which float4 of the row
    if (e >= N_EDGES) return;
    int s = src[e];
    int d = dst[e];
    const float4* hv = (const float4*)(h + (long long)s * FEAT);
    float4 v = hv[q];
    float* o = msum + (long long)d * FEAT + q * 4;
    atomicAdd(o + 0, v.x);
    atomicAdd(o + 1, v.y);
    atomicAdd(o + 2, v.z);
    atomicAdd(o + 3, v.w);
    if (q == 0) atomicAdd(deg + d, 1.0f);
}

// ---------------------------------------------------------------------------
// Kernel 3: out[m,n] = sum_k htot[m,k] * W[n,k] + b[n]
//   htot[m,k] = h[m,k]              (k < 64)
//             = msum[m,k-64]/deg'   (k >= 64), deg' = max(deg[m],1)
// One wave32 computes a 16(M) x 64(N) stripe with V_WMMA_F32_16X16X4_F32.
// A-frag (16x4 f32): lanes 0-15 hold K=0,1 ; lanes 16-31 hold K=2,3.
// B-frag (4x16 f32): mirrored (lane = N, half selects K pair).
// C/D (16x16 f32): v8f, M = vgpr + 8*half, N = lane&15.
// ---------------------------------------------------------------------------
__global__ void __launch_bounds__(128)
sage_gemm_wmma_kernel(const float* __restrict__ h,     // [N,64]
                      const float* __restrict__ W,     // [64,128]
                      const float* __restrict__ bias,  // [64]
                      const float* __restrict__ msum,  // [N,64]
                      const float* __restrict__ deg,   // [N]
                      float* __restrict__ out) {       // [N,64]
    const int wave = threadIdx.x >> 5;
    const int lane = threadIdx.x & 31;
    const int tile = blockIdx.x * 4 + wave;     // 16-row tile index
    const int m0   = tile * 16;
    if (m0 >= N_NODES) return;                  // wave-uniform exit

    const int hi  = lane >> 4;                  // K-pair select (0 or 1)
    const int l16 = lane & 15;                  // row (A) / col (B,C)

    const long long arow = (long long)(m0 + l16);
    const float invd = 1.0f / fmaxf(deg[arow], 1.0f);
    const float* __restrict__ hrow = h    + arow * FEAT;
    const float* __restrict__ mrow = msum + arow * FEAT;
    const float* __restrict__ wrow = W + (long long)l16 * 128; // W[n=l16][*]

    v8f c0 = {}, c1 = {}, c2 = {}, c3 = {};

    // ---- first K-half: A from h ----
#pragma unroll 4
    for (int k = 0; k < 64; k += 4) {
        const int ka = k + (hi << 1);
        v2f a; a.x = hrow[ka]; a.y = hrow[ka + 1];
        const float* wp = wrow + ka;
        v2f b0; b0.x = wp[0];        b0.y = wp[1];
        v2f b1; b1.x = wp[16 * 128]; b1.y = wp[16 * 128 + 1];
        v2f b2; b2.x = wp[32 * 128]; b2.y = wp[32 * 128 + 1];
        v2f b3; b3.x = wp[48 * 128]; b3.y = wp[48 * 128 + 1];
        c0 = __builtin_amdgcn_wmma_f32_16x16x4_f32(false, a, false, b0, (short)0, c0, false, false);
        c1 = __builtin_amdgcn_wmma_f32_16x16x4_f32(false, a, false, b1, (short)0, c1, false, false);
        c2 = __builtin_amdgcn_wmma_f32_16x16x4_f32(false, a, false, b2, (short)0, c2, false, false);
        c3 = __builtin_amdgcn_wmma_f32_16x16x4_f32(false, a, false, b3, (short)0, c3, false, false);
    }
    // ---- second K-half: A from msum * (1/deg) ----
#pragma unroll 4
    for (int k = 64; k < 128; k += 4) {
        const int ka = k + (hi << 1);
        v2f a; a.x = mrow[ka - 64] * invd; a.y = mrow[ka - 63] * invd;
        const float* wp = wrow + ka;
        v2f b0; b0.x = wp[0];        b0.y = wp[1];
        v2f b1; b1.x = wp[16 * 128]; b1.y = wp[16 * 128 + 1];
        v2f b2; b2.x = wp[32 * 128]; b2.y = wp[32 * 128 + 1];
        v2f b3; b3.x = wp[48 * 128]; b3.y = wp[48 * 128 + 1];
        c0 = __builtin_amdgcn_wmma_f32_16x16x4_f32(false, a, false, b0, (short)0, c0, false, false);
        c1 = __builtin_amdgcn_wmma_f32_16x16x4_f32(false, a, false, b1, (short)0, c1, false, false);
        c2 = __builtin_amdgcn_wmma_f32_16x16x4_f32(false, a, false, b2, (short)0, c2, false, false);
        c3 = __builtin_amdgcn_wmma_f32_16x16x4_f32(false, a, false, b3, (short)0, c3, false, false);
    }

    // ---- add bias, store ----
    const float bb0 = bias[l16];
    const float bb1 = bias[16 + l16];
    const float bb2 = bias[32 + l16];
    const float bb3 = bias[48 + l16];
#pragma unroll
    for (int v = 0; v < 8; ++v) {
        const int m = m0 + v + (hi << 3);
        float* orow = out + (long long)m * FEAT;
        orow[l16]      = c0[v] + bb0;
        orow[16 + l16] = c1[v] + bb1;
        orow[32 + l16] = c2[v] + bb2;
        orow[48 + l16] = c3[v] + bb3;
    }
}

// ---------------------------------------------------------------------------
extern "C" void kernel_launch(void* const* d_in, const int* in_sizes, int n_in,
                              void* d_out, int out_size, void* d_ws, size_t ws_size,
                              hipStream_t stream) {
    const float* h    = (const float*)d_in[0];   // [100000, 64]
    const float* W    = (const float*)d_in[1];   // [64, 128]
    const float* bias = (const float*)d_in[2];   // [64]
    const int*   src  = (const int*)d_in[3];     // [1250000]
    const int*   dst  = (const int*)d_in[4];     // [1250000]
    float*       out  = (float*)d_out;           // [100000, 64]

    float* msum = (float*)d_ws;                      // [N, 64]
    float* deg  = msum + (size_t)N_NODES * FEAT;     // [N]

    // 1) zero accumulators
    const int n_zero = N_NODES * FEAT + N_NODES;
    sage_zero_kernel<<<4096, 256, 0, stream>>>(msum, n_zero);

    // 2) edge scatter (16 threads/edge)
    const long long n_thr = (long long)N_EDGES * 16;
    const int blocks_e = (int)((n_thr + 255) / 256);
    sage_edge_scatter_kernel<<<blocks_e, 256, 0, stream>>>(h, src, dst, msum, deg);

    // 3) fused mean + concat + linear via f32 WMMA
    const int n_tiles  = N_NODES / 16;               // 6250 (exact)
    const int blocks_g = (n_tiles + 3) / 4;          // 4 waves/block
    sage_gemm_wmma_kernel<<<blocks_g, 128, 0, stream>>>(h, W, bias, msum, deg, out);
}